// GAE_39822936769088
// MI455X (gfx1250) — compile-verified
//
#include <hip/hip_runtime.h>
#include <cstdint>
#include <cstddef>

// ---------------------------------------------------------------------------
// GAE backward scan, blocked 3-pass formulation for MI455X (gfx1250, wave32).
//   delta_t = r_t + GAMMA*nv_t*nd_t - v_t
//   gae_t   = delta_t + (GAMMA*LMBDA*nd_t) * gae_{t+1}      (affine in carry)
// Segment affine composition: over a segment, gae_first = b + a * carry_in,
// with a = prod(c_t), b = local scan with zero carry.
// ---------------------------------------------------------------------------

typedef float vf4 __attribute__((ext_vector_type(4)));
typedef int   vi4 __attribute__((ext_vector_type(4)));
typedef float vf2 __attribute__((ext_vector_type(2)));

#define GAMMA_F 0.99f
#define GL_F    (0.99f * 0.95f)

constexpr int B_   = 8192;
constexpr int T_   = 2048;
constexpr int SEG  = 64;            // segment length (timesteps per thread)
constexpr int S_   = T_ / SEG;      // 32 segments per row
constexpr int L4   = SEG / 4;       // 16 float4 chunks per segment
constexpr int BLOCK = 256;          // 8 waves (wave32)
constexpr int NSEG_TOT = B_ * S_;   // 262144 threads for passes 1 & 3

// ---- gfx1250 async global->LDS path (guarded; falls back to plain loads) ---
#if defined(__gfx1250__) && __has_builtin(__builtin_amdgcn_global_load_async_to_lds_b128)
#define USE_ASYNC 1
#else
#define USE_ASYNC 0
#endif

#if USE_ASYNC
// Builtin signature (from hipcc diagnostic): pointer params are int4* with
// address-space qualifiers: (v4i addrspace(1)*, v4i addrspace(3)*, Imm, Imm).
typedef __attribute__((address_space(1))) vi4 gvi4_t;
typedef __attribute__((address_space(3))) vi4 lvi4_t;

__device__ __forceinline__ void async_ld16(const void* g, void* l) {
  // copies 16 bytes per lane from the global address g to the LDS address l; tracked by ASYNCcnt
  __builtin_amdgcn_global_load_async_to_lds_b128((gvi4_t*)g, (lvi4_t*)l, 0, 0);
}
__device__ __forceinline__ void wait_async4() {
#if __has_builtin(__builtin_amdgcn_s_wait_asynccnt)
  __builtin_amdgcn_s_wait_asynccnt(4);
  asm volatile("" ::: "memory");      // keep LDS reads below the wait
#else
  asm volatile("s_wait_asynccnt 4" ::: "memory");
#endif
}
__device__ __forceinline__ void wait_async0() {
#if __has_builtin(__builtin_amdgcn_s_wait_asynccnt)
  __builtin_amdgcn_s_wait_asynccnt(0);
  asm volatile("" ::: "memory");
#else
  asm volatile("s_wait_asynccnt 0" ::: "memory");
#endif
}
#endif

// ---------------------------------------------------------------------------
// Pass 1: per-segment affine composition (a, b)
// gid = row*S + seg, so base element index = gid*SEG (segments contiguous:
// one wave spans 32*64*4B = 8KB contiguous per input array).
// ---------------------------------------------------------------------------
__global__ void __launch_bounds__(BLOCK)
gae_partials(const float* __restrict__ reward, const int* __restrict__ term,
             const float* __restrict__ value,  const float* __restrict__ nextv,
             vf2* __restrict__ part)
{
  const int gid = (int)blockIdx.x * BLOCK + (int)threadIdx.x;
  if (gid >= NSEG_TOT) return;
  const size_t base = (size_t)gid * SEG;

  // Prefetch the low (read-last) end of this lane's segment.
  __builtin_prefetch(reward + base, 0, 1);
  __builtin_prefetch(term   + base, 0, 1);
  __builtin_prefetch(value  + base, 0, 1);
  __builtin_prefetch(nextv  + base, 0, 1);

  float a = 1.0f, b = 0.0f;
  for (int j = L4 - 1; j >= 0; --j) {
    const size_t o = base + (size_t)j * 4;
    vf4 rw = *(const vf4*)(reward + o);
    vi4 tm = *(const vi4*)(term   + o);
    vf4 vv = *(const vf4*)(value  + o);
    vf4 nv = *(const vf4*)(nextv  + o);
    float nd, c, d;
    nd = 1.0f - (float)tm.w; c = GL_F * nd; d = fmaf(GAMMA_F * nd, nv.w, rw.w) - vv.w; b = fmaf(c, b, d); a *= c;
    nd = 1.0f - (float)tm.z; c = GL_F * nd; d = fmaf(GAMMA_F * nd, nv.z, rw.z) - vv.z; b = fmaf(c, b, d); a *= c;
    nd = 1.0f - (float)tm.y; c = GL_F * nd; d = fmaf(GAMMA_F * nd, nv.y, rw.y) - vv.y; b = fmaf(c, b, d); a *= c;
    nd = 1.0f - (float)tm.x; c = GL_F * nd; d = fmaf(GAMMA_F * nd, nv.x, rw.x) - vv.x; b = fmaf(c, b, d); a *= c;
  }
  vf2 ab; ab.x = a; ab.y = b;
  part[gid] = ab;
}

// ---------------------------------------------------------------------------
// Pass 2: per-row backward scan over the 32 segment compositions.
// Overwrites part[row*S+s].x with the carry *entering* segment s from the
// right (i.e. gae at the first element of segment s+1). Tiny: 2MB traffic.
// ---------------------------------------------------------------------------
__global__ void __launch_bounds__(BLOCK)
gae_carries(vf2* __restrict__ part)
{
  const int r = (int)blockIdx.x * BLOCK + (int)threadIdx.x;
  if (r >= B_) return;
  float carry = 0.0f;
  int idx = r * S_ + (S_ - 1);
  for (int s = S_ - 1; s >= 0; --s, --idx) {
    vf2 ab = part[idx];
    vf2 w; w.x = carry; w.y = 0.0f;
    part[idx] = w;                       // carry-in for this segment
    carry = fmaf(ab.x, carry, ab.y);     // gae at first element of segment s
  }
}

// ---------------------------------------------------------------------------
// Pass 3: re-run local scan seeded with carry; write advantages & returns.
// Blocks reversed so pass-1's stream tail (hot in 192MB L2) is re-read first.
// Inputs staged through LDS via double-buffered async copies when available.
// ---------------------------------------------------------------------------
__global__ void __launch_bounds__(BLOCK)
gae_apply(const float* __restrict__ reward, const int* __restrict__ term,
          const float* __restrict__ value,  const float* __restrict__ nextv,
          const vf2* __restrict__ part,
          float* __restrict__ adv, float* __restrict__ ret)
{
  const int bid = (int)gridDim.x - 1 - (int)blockIdx.x;   // reversed issue order
  const int gid = bid * BLOCK + (int)threadIdx.x;
  if (gid >= NSEG_TOT) return;
  const size_t base = (size_t)gid * SEG;
  float gae = part[gid].x;

#if USE_ASYNC
  __shared__ vf4 stage[2][4][BLOCK];     // 32 KB: [buf][array][lane] x 16B
  const int tid = (int)threadIdx.x;
  #define ISSUE(buf, j) do {                                   \
      const size_t o_ = base + (size_t)(j) * 4;                \
      async_ld16(reward + o_, &stage[(buf)][0][tid]);          \
      async_ld16(term   + o_, &stage[(buf)][1][tid]);          \
      async_ld16(value  + o_, &stage[(buf)][2][tid]);          \
      async_ld16(nextv  + o_, &stage[(buf)][3][tid]);          \
    } while (0)

  ISSUE(0, L4 - 1);
  int cur = 0;
  for (int j = L4 - 1; j >= 0; --j) {
    if (j > 0) { ISSUE(cur ^ 1, j - 1); wait_async4(); }  // overlap next stage
    else       { wait_async0(); }
    vf4 rw = stage[cur][0][tid];
    vi4 tm = *(const vi4*)&stage[cur][1][tid];
    vf4 vv = stage[cur][2][tid];
    vf4 nv = stage[cur][3][tid];
    cur ^= 1;

    float nd, c, d; vf4 av, rt;
    nd = 1.0f - (float)tm.w; c = GL_F * nd; d = fmaf(GAMMA_F * nd, nv.w, rw.w) - vv.w; gae = fmaf(c, gae, d); av.w = gae; rt.w = gae + vv.w;
    nd = 1.0f - (float)tm.z; c = GL_F * nd; d = fmaf(GAMMA_F * nd, nv.z, rw.z) - vv.z; gae = fmaf(c, gae, d); av.z = gae; rt.z = gae + vv.z;
    nd = 1.0f - (float)tm.y; c = GL_F * nd; d = fmaf(GAMMA_F * nd, nv.y, rw.y) - vv.y; gae = fmaf(c, gae, d); av.y = gae; rt.y = gae + vv.y;
    nd = 1.0f - (float)tm.x; c = GL_F * nd; d = fmaf(GAMMA_F * nd, nv.x, rw.x) - vv.x; gae = fmaf(c, gae, d); av.x = gae; rt.x = gae + vv.x;

    const size_t o = base + (size_t)j * 4;
    __builtin_nontemporal_store(av, (vf4*)(adv + o));
    __builtin_nontemporal_store(rt, (vf4*)(ret + o));
  }
  #undef ISSUE
#else
  for (int j = L4 - 1; j >= 0; --j) {
    const size_t o = base + (size_t)j * 4;
    vf4 rw = *(const vf4*)(reward + o);
    vi4 tm = *(const vi4*)(term   + o);
    vf4 vv = *(const vf4*)(value  + o);
    vf4 nv = *(const vf4*)(nextv  + o);

    float nd, c, d; vf4 av, rt;
    nd = 1.0f - (float)tm.w; c = GL_F * nd; d = fmaf(GAMMA_F * nd, nv.w, rw.w) - vv.w; gae = fmaf(c, gae, d); av.w = gae; rt.w = gae + vv.w;
    nd = 1.0f - (float)tm.z; c = GL_F * nd; d = fmaf(GAMMA_F * nd, nv.z, rw.z) - vv.z; gae = fmaf(c, gae, d); av.z = gae; rt.z = gae + vv.z;
    nd = 1.0f - (float)tm.y; c = GL_F * nd; d = fmaf(GAMMA_F * nd, nv.y, rw.y) - vv.y; gae = fmaf(c, gae, d); av.y = gae; rt.y = gae + vv.y;
    nd = 1.0f - (float)tm.x; c = GL_F * nd; d = fmaf(GAMMA_F * nd, nv.x, rw.x) - vv.x; gae = fmaf(c, gae, d); av.x = gae; rt.x = gae + vv.x;

    __builtin_nontemporal_store(av, (vf4*)(adv + o));
    __builtin_nontemporal_store(rt, (vf4*)(ret + o));
  }
#endif
}

// ---------------------------------------------------------------------------
extern "C" void kernel_launch(void* const* d_in, const int* in_sizes, int n_in,
                              void* d_out, int out_size, void* d_ws, size_t ws_size,
                              hipStream_t stream)
{
  (void)in_sizes; (void)n_in; (void)out_size; (void)ws_size;

  const float* reward = (const float*)d_in[0];
  const int*   term   = (const int*)  d_in[1];
  const float* value  = (const float*)d_in[2];
  const float* nextv  = (const float*)d_in[3];

  float* adv = (float*)d_out;                 // advantages [B,T]
  float* ret = adv + (size_t)B_ * T_;         // returns    [B,T]
  vf2*   part = (vf2*)d_ws;                   // B*S float2 = 2 MiB scratch

  const dim3 blk(BLOCK);
  const dim3 grdA((NSEG_TOT + BLOCK - 1) / BLOCK);   // 1024 blocks = 8192 waves
  const dim3 grdB((B_ + BLOCK - 1) / BLOCK);         // 32 blocks

  gae_partials<<<grdA, blk, 0, stream>>>(reward, term, value, nextv, part);
  gae_carries <<<grdB, blk, 0, stream>>>(part);
  gae_apply   <<<grdA, blk, 0, stream>>>(reward, term, value, nextv, part, adv, ret);
}